// TemporalGNN_67980742361881
// MI455X (gfx1250) — compile-verified
//
#include <hip/hip_runtime.h>
#include <hip/hip_bf16.h>
#include <math.h>

typedef __attribute__((ext_vector_type(2))) float v2f;
typedef __attribute__((ext_vector_type(8))) float v8f;

#define NN 10000
#define FFEAT 4
#define TPER 137
#define EDGES 160000
#define DEMB 32
#define FT (FFEAT * TPER)        // 548
#define ETOT (EDGES + NN)        // 170000 (edges + self loops)

// Workspace layout (float offsets)
#define WS_AGG   0                       // N*F*T = 5,480,000
#define WS_DEG   5480000                 // N (deg, then dinv in place)
#define WS_PROBS 5490000                 // 144 (padded, pad zeroed)
#define WS_WEFF  5490144                 // 2*4*32 = 256 (gates 0 and 2)
#define WS_BEFF  5490400                 // 2*32 = 64
#define WS_NORM  5490464                 // ETOT
#define WS_SRC   5660464                 // ETOT (int)
#define WS_DST   5830464                 // ETOT (int)

#if __has_builtin(__builtin_amdgcn_tanhf)
__device__ __forceinline__ float fast_tanh(float x) { return __builtin_amdgcn_tanhf(x); }
#else
__device__ __forceinline__ float fast_tanh(float x) { return tanhf(x); }
#endif

// ---------------- zero agg + deg (float4 vectorized) ----------------
__global__ void k_zero(float* ws) {
    float4 z = make_float4(0.f, 0.f, 0.f, 0.f);
    float4* agg4 = (float4*)(ws + WS_AGG);
    float4* deg4 = (float4*)(ws + WS_DEG);
    size_t stride = (size_t)gridDim.x * blockDim.x;
    size_t i = (size_t)blockIdx.x * blockDim.x + threadIdx.x;
    for (; i < (size_t)(NN * FT / 4); i += stride) agg4[i] = z;
    i = (size_t)blockIdx.x * blockDim.x + threadIdx.x;
    for (; i < (size_t)(NN / 4); i += stride) deg4[i] = z;
}

// ---------------- degree accumulation ----------------
__global__ void k_deg(const int* __restrict__ ei, float* __restrict__ deg) {
    int e = blockIdx.x * blockDim.x + threadIdx.x;
    if (e < EDGES) atomicAdd(&deg[ei[EDGES + e]], 1.0f);   // col = target
}

// ---------------- dinv = rsqrt(deg + 1 self loop), in place ----------------
__global__ void k_dinv(float* __restrict__ deg) {
    int n = blockIdx.x * blockDim.x + threadIdx.x;
    if (n < NN) deg[n] = rsqrtf(deg[n] + 1.0f);
}

// ---------------- softmax(attention), Weff/beff contraction, out init ----------------
__global__ void k_small(const float* __restrict__ att,
                        const float* __restrict__ conv_w, const float* __restrict__ conv_b,
                        const float* __restrict__ lin_w,  const float* __restrict__ lin_b,
                        const float* __restrict__ b2,
                        float* __restrict__ probs, float* __restrict__ weff,
                        float* __restrict__ beff,  float* __restrict__ out) {
    __shared__ float sh[256];
    int tid = threadIdx.x;
    float a = (tid < TPER) ? att[tid] : -1e30f;
    sh[tid] = a; __syncthreads();
    for (int s = 128; s > 0; s >>= 1) {
        if (tid < s) sh[tid] = fmaxf(sh[tid], sh[tid + s]);
        __syncthreads();
    }
    float mx = sh[0]; __syncthreads();
    float ex = (tid < TPER) ? __expf(a - mx) : 0.0f;
    sh[tid] = ex; __syncthreads();
    for (int s = 128; s > 0; s >>= 1) {
        if (tid < s) sh[tid] += sh[tid + s];
        __syncthreads();
    }
    float inv = 1.0f / sh[0];
    if (tid < TPER) probs[tid] = ex * inv;
    if (tid >= TPER && tid < 144) probs[tid] = 0.0f;   // zero pad: gate kernel loads unguarded

    // Weff[gi][f][d] = sum_k conv_w[g,f,k] * lin_w[g,k,d], g = gi*2 (gates z and h)
    if (tid < 2 * FFEAT * DEMB) {
        int gi = tid >> 7, f = (tid >> 5) & 3, d = tid & 31, g = gi * 2;
        float s = 0.0f;
        for (int k = 0; k < DEMB; ++k)
            s += conv_w[g * FFEAT * DEMB + f * DEMB + k] * lin_w[g * 2 * DEMB * DEMB + k * DEMB + d];
        weff[tid] = s;
    }
    // beff[gi][d] = conv_b[g,:] @ lin_w[g,:D,d] + lin_b[g,d]
    if (tid < 2 * DEMB) {
        int gi = tid >> 5, d = tid & 31, g = gi * 2;
        float s = lin_b[g * DEMB + d];
        for (int k = 0; k < DEMB; ++k)
            s += conv_b[g * DEMB + k] * lin_w[g * 2 * DEMB * DEMB + k * DEMB + d];
        beff[tid] = s;
    }
    if (tid == 0) out[0] = b2[0];
}

// ---------------- per-edge norm + src/dst (incl. self loops) ----------------
__global__ void k_norm(const int* __restrict__ ei, const float* __restrict__ dinv,
                       float* __restrict__ nrm, int* __restrict__ src, int* __restrict__ dst) {
    int e = blockIdx.x * blockDim.x + threadIdx.x;
    if (e >= ETOT) return;
    int r, c;
    if (e < EDGES) { r = ei[e]; c = ei[EDGES + e]; }
    else           { r = c = e - EDGES; }
    src[e] = r;
    dst[e] = c;
    nrm[e] = dinv[r] * dinv[c];
}

// ---------------- scatter-add aggregation: agg[c,f,t] += norm * x[r,f,t] ----------------
// float4 per thread: 16B coalesced gather + 4 f32 atomics.
__global__ void k_agg(const float* __restrict__ x, const float* __restrict__ nrm,
                      const int* __restrict__ src, const int* __restrict__ dst,
                      float* __restrict__ agg) {
    unsigned idx = blockIdx.x * blockDim.x + threadIdx.x;
    if (idx >= (unsigned)ETOT * (FT / 4)) return;
    unsigned e = idx / (FT / 4);
    unsigned j = (idx - e * (FT / 4)) * 4;
    float nv = nrm[e];
    const float4 v = *(const float4*)(x + (size_t)src[e] * FT + j);
    float* d = agg + (size_t)dst[e] * FT + j;
    atomicAdd(d + 0, v.x * nv);
    atomicAdd(d + 1, v.y * nv);
    atomicAdd(d + 2, v.z * nv);
    atomicAdd(d + 3, v.w * nv);
}

// ---------------- WMMA gate + attention pool + head, one wave per node ----------------
__global__ void __launch_bounds__(32)
k_gate(const float* __restrict__ agg, const float* __restrict__ probs,
       const float* __restrict__ weff, const float* __restrict__ beff,
       const float* __restrict__ w1, const float* __restrict__ b1,
       const float* __restrict__ w2, float* __restrict__ out) {
    int n = blockIdx.x;
    int lane = threadIdx.x;
    int mrow = lane & 15;
    int kb = (lane >> 4) * 2;      // K pair this lane-half holds (0 or 2)

    // B operands: Weff is [gi][k(=F)][d]; B is 4x16 (KxN), N striped over lane, K over (vgpr, lane-half)
    v2f b0lo, b0hi, b2lo, b2hi;
    b0lo.x = weff[0 * 128 + kb * 32 + mrow];
    b0lo.y = weff[0 * 128 + (kb + 1) * 32 + mrow];
    b0hi.x = weff[0 * 128 + kb * 32 + mrow + 16];
    b0hi.y = weff[0 * 128 + (kb + 1) * 32 + mrow + 16];
    b2lo.x = weff[128 + kb * 32 + mrow];
    b2lo.y = weff[128 + (kb + 1) * 32 + mrow];
    b2hi.x = weff[128 + kb * 32 + mrow + 16];
    b2hi.y = weff[128 + (kb + 1) * 32 + mrow + 16];

    // Bias folded into C accumulator (bias depends only on output column N)
    float be0lo = beff[mrow], be0hi = beff[mrow + 16];
    float be2lo = beff[32 + mrow], be2hi = beff[32 + mrow + 16];

    float accLo = 0.0f, accHi = 0.0f;
    const float* ab = agg + (size_t)n * FT;
    int mb = (lane >> 4) * 8;      // C-layout M base for this lane-half

    for (int t0 = 0; t0 < TPER; t0 += 16) {
        int t = t0 + mrow;
        // Unguarded loads: overreads land in always-initialized finite ws regions;
        // out-of-range rows are masked by probs pad == 0.
        v2f a;
        a.x = ab[kb * TPER + t];
        a.y = ab[(kb + 1) * TPER + t];

        v8f c0lo = {be0lo, be0lo, be0lo, be0lo, be0lo, be0lo, be0lo, be0lo};
        v8f c0hi = {be0hi, be0hi, be0hi, be0hi, be0hi, be0hi, be0hi, be0hi};
        v8f c2lo = {be2lo, be2lo, be2lo, be2lo, be2lo, be2lo, be2lo, be2lo};
        v8f c2hi = {be2hi, be2hi, be2hi, be2hi, be2hi, be2hi, be2hi, be2hi};

        c0lo = __builtin_amdgcn_wmma_f32_16x16x4_f32(false, a, false, b0lo, (short)0, c0lo, false, false);
        c0hi = __builtin_amdgcn_wmma_f32_16x16x4_f32(false, a, false, b0hi, (short)0, c0hi, false, false);
        c2lo = __builtin_amdgcn_wmma_f32_16x16x4_f32(false, a, false, b2lo, (short)0, c2lo, false, false);
        c2hi = __builtin_amdgcn_wmma_f32_16x16x4_f32(false, a, false, b2hi, (short)0, c2hi, false, false);

        #pragma unroll
        for (int v = 0; v < 8; ++v) {
            float p = probs[t0 + mb + v];
            // Hout = (1 - sigmoid(g0)) * tanh(g2);  1 - sigmoid(z) = 0.5*(1 - tanh(z/2))
            float s0 = 0.5f - 0.5f * fast_tanh(0.5f * c0lo[v]);
            accLo += p * s0 * fast_tanh(c2lo[v]);
            float s1 = 0.5f - 0.5f * fast_tanh(0.5f * c0hi[v]);
            accHi += p * s1 * fast_tanh(c2hi[v]);
        }
    }

    // fold the two M-halves (lane L and L+16 hold same output column d)
    accLo += __shfl_xor(accLo, 16, 32);
    accHi += __shfl_xor(accHi, 16, 32);

    // head: y[n] = relu(h[n]) . w1 + b1 ; out += w2[n] * y[n]
    float dot = 0.0f;
    if (lane < 16)
        dot = w1[lane] * fmaxf(accLo, 0.0f) + w1[lane + 16] * fmaxf(accHi, 0.0f);
    for (int off = 8; off >= 1; off >>= 1) dot += __shfl_xor(dot, off, 32);
    if (lane == 0) atomicAdd(out, w2[n] * (dot + b1[0]));
}

extern "C" void kernel_launch(void* const* d_in, const int* in_sizes, int n_in,
                              void* d_out, int out_size, void* d_ws, size_t ws_size,
                              hipStream_t stream) {
    const float* x      = (const float*)d_in[0];
    const int*   ei     = (const int*)d_in[1];
    const float* conv_w = (const float*)d_in[2];
    const float* conv_b = (const float*)d_in[3];
    const float* lin_w  = (const float*)d_in[4];
    const float* lin_b  = (const float*)d_in[5];
    const float* att    = (const float*)d_in[6];
    const float* w1     = (const float*)d_in[7];
    const float* b1     = (const float*)d_in[8];
    const float* w2     = (const float*)d_in[9];
    const float* b2     = (const float*)d_in[10];
    float* out = (float*)d_out;

    float* ws    = (float*)d_ws;
    float* agg   = ws + WS_AGG;
    float* deg   = ws + WS_DEG;     // becomes dinv
    float* probs = ws + WS_PROBS;
    float* weff  = ws + WS_WEFF;
    float* beff  = ws + WS_BEFF;
    float* nrm   = ws + WS_NORM;
    int*   src   = (int*)(ws + WS_SRC);
    int*   dst   = (int*)(ws + WS_DST);

    k_zero<<<4096, 256, 0, stream>>>(ws);
    k_deg<<<(EDGES + 255) / 256, 256, 0, stream>>>(ei, deg);
    k_dinv<<<(NN + 255) / 256, 256, 0, stream>>>(deg);
    k_small<<<1, 256, 0, stream>>>(att, conv_w, conv_b, lin_w, lin_b, b2,
                                   probs, weff, beff, out);
    k_norm<<<(ETOT + 255) / 256, 256, 0, stream>>>(ei, deg, nrm, src, dst);
    {
        unsigned total = (unsigned)ETOT * (FT / 4);
        k_agg<<<(total + 255u) / 256u, 256, 0, stream>>>(x, nrm, src, dst, agg);
    }
    k_gate<<<NN, 32, 0, stream>>>(agg, probs, weff, beff, w1, b1, w2, out);
}